// OnlineConceptCompressionModel_5446018531662
// MI455X (gfx1250) — compile-verified
//
#include <hip/hip_runtime.h>
#include <math.h>

// =====================================================================
// OnlineConceptCompressionModel forward for MI455X (gfx1250, wave32).
// All dense projections run through an LDS-tiled f32 WMMA GEMM
// (V_WMMA_F32_16X16X4_F32): 128x128 block tile, 8 waves, each wave owns
// a 32x64 sub-tile (8 v8f accumulators) -> 32 WMMAs per K-tile against
// 48 LDS fragment reads (1.5 ds-reads/WMMA). Compute-bound workload
// (~540 GFLOP of GEMM, weights reused 190x in the deep stage).
// f32 WMMA chosen over bf16 16x16x32 to stay bit-faithful to the f32
// reference.
// Workspace use: ~435 MB of d_ws.
// =====================================================================

typedef float v2f __attribute__((ext_vector_type(2)));
typedef float v8f __attribute__((ext_vector_type(8)));

#define B_    2
#define L_    96
#define H_    768
#define NH_   12
#define HD_   64
#define FF_   2048
#define V_    32000
#define K_    512
#define T_    95          // L-1
#define N_    190         // B*T
#define SMAX_ 96
#define VERY_NEG_ (-10000.0f)

// ------------------------- generic WMMA GEMM -------------------------
enum { EPI_NONE = 0, EPI_BIAS, EPI_ADD, EPI_BIAS_ADD, EPI_GELU_BIAS,
       EPI_SILU_MUL, EPI_ROWSCALE };

__device__ __forceinline__ float epi_apply(float v, int m, int n, int ldc,
                                           int epi, const float* bias,
                                           const float* aux) {
  switch (epi) {
    case EPI_BIAS:     return v + bias[n];
    case EPI_ADD:      return v + aux[(size_t)m * ldc + n];
    case EPI_BIAS_ADD: return v + bias[n] + aux[(size_t)m * ldc + n];
    case EPI_GELU_BIAS: {
      float x = v + bias[n];
      return 0.5f * x * (1.0f + erff(x * 0.70710678118654752f));  // exact gelu
    }
    case EPI_SILU_MUL: {
      float g = aux[(size_t)m * ldc + n];           // pre-stored gate
      return (g / (1.0f + expf(-g))) * v;           // silu(g)*u
    }
    case EPI_ROWSCALE: return v * aux[m];
    default:           return v;
  }
}

#define BM_ 128
#define BN_ 128
#define BK_ 16

// C[M,N] = A[M,K] @ B[K,N] (+epilogue). Fully bounds-checked.
// 8 waves in a 4(row)x2(col) grid; each wave: 2 M-frags x 4 N-frags.
__global__ __launch_bounds__(256)
void k_gemm(const float* __restrict__ A, int lda,
            const float* __restrict__ Bm, int ldb,
            float* __restrict__ C, int ldc,
            const float* __restrict__ bias, const float* __restrict__ aux,
            int M, int N, int Kd, int epi) {
  __shared__ float As[BM_][BK_ + 1];                 // pad: stride 17 banks
  __shared__ float Bs[BK_][BN_ + 4];                 // pad vs bank conflicts
  const int tid  = threadIdx.x;
  const int lane = tid & 31;
  const int wave = tid >> 5;                         // 0..7
  const int wrow = wave & 3;                         // 4 row groups of 32
  const int wcol = wave >> 2;                        // 2 col groups of 64
  const int m0 = blockIdx.x * BM_;
  const int n0 = blockIdx.y * BN_;
  const v8f vz = {0, 0, 0, 0, 0, 0, 0, 0};
  v8f acc[2][4];
#pragma unroll
  for (int i = 0; i < 2; ++i)
#pragma unroll
    for (int j = 0; j < 4; ++j) acc[i][j] = vz;
  const int mf = lane & 15;                          // M (or N) index in frag
  const int kb = (lane >> 4) << 1;                   // K sub-offset 0 / 2

  for (int k0 = 0; k0 < Kd; k0 += BK_) {
#pragma unroll
    for (int i = 0; i < 8; ++i) {                    // A tile: 128x16
      int e = tid + i * 256;
      int r = e >> 4, c = e & 15;
      int gm = m0 + r, gk = k0 + c;
      As[r][c] = (gm < M && gk < Kd) ? A[(size_t)gm * lda + gk] : 0.0f;
    }
#pragma unroll
    for (int i = 0; i < 8; ++i) {                    // B tile: 16x128
      int e = tid + i * 256;
      int r = e >> 7, c = e & 127;
      int gk = k0 + r, gn = n0 + c;
      Bs[r][c] = (gk < Kd && gn < N) ? Bm[(size_t)gk * ldb + gn] : 0.0f;
    }
    __syncthreads();
#pragma unroll
    for (int kk = 0; kk < BK_; kk += 4) {
      v2f a[2], b[4];
#pragma unroll
      for (int i = 0; i < 2; ++i) {
        // A frag (16x4 f32): lane m = lane%16, VGPR j holds K = kb+j
        a[i].x = As[wrow * 32 + i * 16 + mf][kk + kb];
        a[i].y = As[wrow * 32 + i * 16 + mf][kk + kb + 1];
      }
#pragma unroll
      for (int j = 0; j < 4; ++j) {
        // B frag (4x16 f32): lane n = lane%16, VGPR j holds K = kb+j
        b[j].x = Bs[kk + kb][wcol * 64 + j * 16 + mf];
        b[j].y = Bs[kk + kb + 1][wcol * 64 + j * 16 + mf];
      }
#pragma unroll
      for (int i = 0; i < 2; ++i)
#pragma unroll
        for (int j = 0; j < 4; ++j)
          acc[i][j] = __builtin_amdgcn_wmma_f32_16x16x4_f32(
              false, a[i], false, b[j], (short)0, acc[i][j], false, false);
    }
    __syncthreads();
  }
  // C/D layout: VGPR r -> M = r + 8*(lane/16), N = lane%16
#pragma unroll
  for (int i = 0; i < 2; ++i) {
    int mb = m0 + wrow * 32 + i * 16 + (lane >> 4) * 8;
#pragma unroll
    for (int j = 0; j < 4; ++j) {
      int nc = n0 + wcol * 64 + j * 16 + mf;
      if (nc < N) {
#pragma unroll
        for (int r = 0; r < 8; ++r) {
          int gm = mb + r;
          if (gm < M)
            C[(size_t)gm * ldc + nc] =
                epi_apply(acc[i][j][r], gm, nc, ldc, epi, bias, aux);
        }
      }
    }
  }
}

// ------------------------- elementwise kernels -----------------------
__global__ void k_embed(const int* __restrict__ ids,
                        const float* __restrict__ emb,
                        float* __restrict__ out) {
  int row = blockIdx.x;
  int id = ids[row];
  for (int d = threadIdx.x; d < H_; d += blockDim.x)
    out[(size_t)row * H_ + d] = emb[(size_t)id * H_ + d];
}

__global__ void k_init(const int* __restrict__ amask,
                       int* __restrict__ lengths, int* __restrict__ pos_sh) {
  int t = threadIdx.x;
  if (t < B_ * L_) pos_sh[t] = t % L_;
  if (t < B_) {
    int s = 0;
    for (int l = 0; l < L_; ++l) s += amask[t * L_ + l];
    lengths[t] = s;
  }
}

__global__ __launch_bounds__(256)
void k_rmsnorm(const float* __restrict__ X, const float* __restrict__ W,
               float* __restrict__ Y) {
  int m = blockIdx.x;
  __shared__ float red[256];
  const float* x = X + (size_t)m * H_;
  float s = 0.0f;
  for (int d = threadIdx.x; d < H_; d += 256) { float v = x[d]; s += v * v; }
  red[threadIdx.x] = s;
  __syncthreads();
  for (int o = 128; o > 0; o >>= 1) {
    if (threadIdx.x < o) red[threadIdx.x] += red[threadIdx.x + o];
    __syncthreads();
  }
  float r = rsqrtf(red[0] / (float)H_ + 1e-6f);
  for (int d = threadIdx.x; d < H_; d += 256)
    Y[(size_t)m * H_ + d] = x[d] * W[d] * r;
}

__global__ __launch_bounds__(384)
void k_rope(float* __restrict__ X, const int* __restrict__ pos) {
  int m = blockIdx.x;
  int p = pos[m];
  int idx = threadIdx.x;                    // 384 = NH_*32
  int head = idx >> 5, d = idx & 31;
  float inv = powf(10000.0f, -(float)d / 32.0f);
  float ang = (float)p * inv;
  float c = cosf(ang), s = sinf(ang);
  float* xb = X + (size_t)m * H_ + head * HD_;
  float x1 = xb[d], x2 = xb[d + 32];
  xb[d]      = x1 * c - x2 * s;
  xb[d + 32] = x2 * c + x1 * s;
}

// one block = (sequence, head); S = 96. Causal + prefix-length mask.
__global__ __launch_bounds__(96)
void k_attn(const float* __restrict__ Q, const float* __restrict__ Kb,
            const float* __restrict__ Vb, float* __restrict__ O,
            const int* __restrict__ lens) {
  __shared__ float sm[96 * 64];             // K tile, then reused for V tile
  __shared__ float att[96][96];
  int seq = blockIdx.x, h = blockIdx.y;
  int len = lens[seq];
  int i = threadIdx.x;                      // query row 0..95
  int base = seq * SMAX_;
  for (int d = 0; d < HD_; ++d)             // load K rows
    sm[i * HD_ + d] = Kb[(size_t)(base + i) * H_ + h * HD_ + d];
  float qreg[HD_];
#pragma unroll
  for (int d = 0; d < HD_; ++d)
    qreg[d] = Q[(size_t)(base + i) * H_ + h * HD_ + d];
  __syncthreads();
  for (int j = 0; j < SMAX_; ++j) {
    float s = 0.0f;
#pragma unroll
    for (int d = 0; d < HD_; ++d) s += qreg[d] * sm[j * HD_ + d];
    bool allow = (j <= i) && (j < len) && (i < len);
    att[i][j] = s * 0.125f + (allow ? 0.0f : VERY_NEG_);
  }
  __syncthreads();                          // all K reads done
  for (int d = 0; d < HD_; ++d)             // overwrite with V rows
    sm[i * HD_ + d] = Vb[(size_t)(base + i) * H_ + h * HD_ + d];
  // softmax over own row (independent of sm)
  float mx = -INFINITY;
  for (int j = 0; j < SMAX_; ++j) mx = fmaxf(mx, att[i][j]);
  float sum = 0.0f;
  for (int j = 0; j < SMAX_; ++j) {
    float e = expf(att[i][j] - mx);
    att[i][j] = e;
    sum += e;
  }
  float inv = 1.0f / sum;
  for (int j = 0; j < SMAX_; ++j) att[i][j] *= inv;
  __syncthreads();                          // V tile ready
  for (int d = 0; d < HD_; ++d) {
    float acc = 0.0f;
    for (int j = 0; j < SMAX_; ++j) acc += att[i][j] * sm[j * HD_ + d];
    O[(size_t)(base + i) * H_ + h * HD_ + d] = acc;
  }
}

// gumbel softmax over 513, argmax (first-max tie rule), is_comp flag
__global__ __launch_bounds__(256)
void k_concept(const float* __restrict__ lg, const float* __restrict__ gum,
               const int* __restrict__ amask, float* __restrict__ Z,
               int* __restrict__ chosen, int* __restrict__ iscomp) {
  int r = blockIdx.x;
  const float* l = lg + (size_t)r * (K_ + 1);
  const float* g = gum + (size_t)r * (K_ + 1);
  __shared__ float rv[256];
  __shared__ int ri[256];
  int tid = threadIdx.x;
  float mx = -INFINITY; int mi = K_ + 1;
  for (int k = tid; k < K_ + 1; k += 256) {
    float y = l[k] + g[k];                       // TAU = 1
    if (y > mx) { mx = y; mi = k; }
  }
  rv[tid] = mx; ri[tid] = mi;
  __syncthreads();
  for (int o = 128; o > 0; o >>= 1) {
    if (tid < o) {
      if (rv[tid + o] > rv[tid] ||
          (rv[tid + o] == rv[tid] && ri[tid + o] < ri[tid])) {
        rv[tid] = rv[tid + o]; ri[tid] = ri[tid + o];
      }
    }
    __syncthreads();
  }
  float gmax = rv[0]; int gidx = ri[0];
  __syncthreads();
  float s = 0.0f;
  for (int k = tid; k < K_ + 1; k += 256) {
    float e = expf(l[k] + g[k] - gmax);
    Z[(size_t)r * (K_ + 1) + k] = e;
    s += e;
  }
  rv[tid] = s;
  __syncthreads();
  for (int o = 128; o > 0; o >>= 1) {
    if (tid < o) rv[tid] += rv[tid + o];
    __syncthreads();
  }
  float inv = 1.0f / rv[0];
  for (int k = tid; k < K_ + 1; k += 256) Z[(size_t)r * (K_ + 1) + k] *= inv;
  if (tid == 0) {
    chosen[r] = gidx;
    iscomp[r] = (gidx != K_) && (amask[r] != 0);
  }
}

// serial per-batch scan: compaction list, cumsum, cummax
__global__ void k_scan(const int* __restrict__ iscomp, int* __restrict__ counts,
                       int* __restrict__ cpos, int* __restrict__ cum,
                       int* __restrict__ latest) {
  int b = threadIdx.x;
  if (b >= B_) return;
  for (int l = 0; l < L_; ++l) cpos[b * L_ + l] = 0;
  int c = 0, lat = -1;
  for (int l = 0; l < L_; ++l) {
    if (iscomp[b * L_ + l]) { cpos[b * L_ + c] = l; c++; lat = l; }
    cum[b * L_ + l] = c;
    latest[b * L_ + l] = lat;
  }
  counts[b] = c;
}

// c_in = compacted hard concept embeddings (ce_all forward == e_hard*validf)
__global__ void k_cin(const int* __restrict__ counts, const int* __restrict__ cpos,
                      const int* __restrict__ chosen,
                      const float* __restrict__ cemb, float* __restrict__ hm) {
  int blk = blockIdx.x, b = blk / L_, slot = blk % L_;
  bool v = slot < counts[b];
  int p = v ? cpos[b * L_ + slot] : 0;
  int ch = v ? chosen[b * L_ + p] : 0;
  for (int d = threadIdx.x; d < H_; d += blockDim.x)
    hm[(size_t)blk * H_ + d] = v ? cemb[(size_t)ch * H_ + d] : 0.0f;
}

__global__ void k_maskmid(float* __restrict__ hm, const int* __restrict__ counts) {
  int blk = blockIdx.x, b = blk / L_, slot = blk % L_;
  float v = (slot < counts[b]) ? 1.0f : 0.0f;
  for (int d = threadIdx.x; d < H_; d += blockDim.x)
    hm[(size_t)blk * H_ + d] *= v;
}

// per-row (e_soft - e_hard)^2 sum for commit loss
__global__ __launch_bounds__(256)
void k_commit(const float* __restrict__ esoft, const int* __restrict__ chosen,
              const int* __restrict__ iscomp, const float* __restrict__ cemb,
              float* __restrict__ out) {
  int r = blockIdx.x;
  int ic = iscomp[r];
  int ch = chosen[r]; if (ch > K_ - 1) ch = K_ - 1;
  __shared__ float red[256];
  float s = 0.0f;
  for (int d = threadIdx.x; d < H_; d += 256) {
    float eh = ic ? cemb[(size_t)ch * H_ + d] : 0.0f;
    float dl = esoft[(size_t)r * H_ + d] - eh;
    s += dl * dl;
  }
  red[threadIdx.x] = s;
  __syncthreads();
  for (int o = 128; o > 0; o >>= 1) {
    if (threadIdx.x < o) red[threadIdx.x] += red[threadIdx.x + o];
    __syncthreads();
  }
  if (threadIdx.x == 0) out[r] = red[0];
}

__global__ void k_hist(const float* __restrict__ Z, const int* __restrict__ amask,
                       float* __restrict__ hist) {
  int k = blockIdx.x * blockDim.x + threadIdx.x;
  if (k >= K_) return;
  float s = 0.0f;
  for (int r = 0; r < B_ * L_; ++r)
    s += Z[(size_t)r * (K_ + 1) + k] * (float)amask[r];
  hist[k] = s;
}

// build deep-stage sequences (seq_h, seq_pos, seq_len, targets, active)
__global__ __launch_bounds__(256)
void k_assemble(const int* __restrict__ lengths, const int* __restrict__ cum,
                const int* __restrict__ latest, const int* __restrict__ cpos,
                const float* __restrict__ cmid, const float* __restrict__ tailo,
                const int* __restrict__ ids, float* __restrict__ seqh,
                int* __restrict__ seqpos, int* __restrict__ seqlen,
                int* __restrict__ lastidx, int* __restrict__ activeI,
                float* __restrict__ activeF, int* __restrict__ targets) {
  int blk = blockIdx.x;
  int n = blk / SMAX_, s = blk % SMAX_;
  int b = n / T_, at = n % T_;
  int cn = cum[b * L_ + at];
  int tau = latest[b * L_ + at];
  int sl = cn + at - tau;                        // cn + (at - tail_start + 1)
  if (sl < 1) sl = 1;
  if (sl > SMAX_) sl = SMAX_;
  bool isc = s < cn;
  int tsrc = tau + 1 + s - cn;
  if (tsrc < 0) tsrc = 0;
  if (tsrc > L_ - 1) tsrc = L_ - 1;
  float vp = (s < sl) ? 1.0f : 0.0f;
  const float* src = isc ? (cmid + (size_t)(b * L_ + s) * H_)
                         : (tailo + (size_t)(b * L_ + tsrc) * H_);
  for (int d = threadIdx.x; d < H_; d += 256)
    seqh[(size_t)blk * H_ + d] = src[d] * vp;
  if (threadIdx.x == 0) {
    seqpos[blk] = isc ? cpos[b * L_ + s] : tsrc;
    if (s == 0) {
      seqlen[n] = sl;
      lastidx[n] = sl - 1;
      int act = (lengths[b] > at + 1) ? 1 : 0;
      activeI[n] = act;
      activeF[n] = (float)act;
      int tt = at + 1; if (tt > L_ - 1) tt = L_ - 1;
      targets[n] = ids[b * L_ + tt];
    }
  }
}

__global__ void k_gather_last(const float* __restrict__ hdn,
                              const int* __restrict__ lastidx,
                              float* __restrict__ lasth) {
  int n = blockIdx.x;
  int row = n * SMAX_ + lastidx[n];
  for (int d = threadIdx.x; d < H_; d += blockDim.x)
    lasth[(size_t)n * H_ + d] = hdn[(size_t)row * H_ + d];
}

__global__ __launch_bounds__(256)
void k_rowloss(const float* __restrict__ logits, const int* __restrict__ targets,
               const int* __restrict__ activeI, float* __restrict__ rowloss) {
  int n = blockIdx.x;
  __shared__ float red[256];
  const float* lr = logits + (size_t)n * V_;
  float mx = -INFINITY;
  for (int k = threadIdx.x; k < V_; k += 256) mx = fmaxf(mx, lr[k]);
  red[threadIdx.x] = mx;
  __syncthreads();
  for (int o = 128; o > 0; o >>= 1) {
    if (threadIdx.x < o)
      red[threadIdx.x] = fmaxf(red[threadIdx.x], red[threadIdx.x + o]);
    __syncthreads();
  }
  mx = red[0];
  __syncthreads();
  float s = 0.0f;
  for (int k = threadIdx.x; k < V_; k += 256) s += expf(lr[k] - mx);
  red[threadIdx.x] = s;
  __syncthreads();
  for (int o = 128; o > 0; o >>= 1) {
    if (threadIdx.x < o) red[threadIdx.x] += red[threadIdx.x + o];
    __syncthreads();
  }
  if (threadIdx.x == 0) {
    float lse = mx + logf(red[0]);
    rowloss[n] = activeI[n] ? (lse - lr[targets[n]]) : 0.0f;
  }
}

// deterministic serial final reduction -> total loss
__global__ void k_total(const float* __restrict__ rowloss,
                        const int* __restrict__ activeI,
                        const float* __restrict__ commit_rows,
                        const float* __restrict__ hist,
                        const float* __restrict__ Z,
                        const int* __restrict__ amask,
                        float* __restrict__ out_total) {
  if (threadIdx.x != 0 || blockIdx.x != 0) return;
  float rec = 0.0f; int na = 0;
  for (int n = 0; n < N_; ++n) { rec += rowloss[n]; na += activeI[n]; }
  rec /= (float)(na < 1 ? 1 : na);
  float cm = 0.0f;
  for (int r = 0; r < B_ * L_; ++r) cm += commit_rows[r];
  cm = 1.5f * cm / (float)(B_ * L_ * H_);          // (1 + BETA_COMMIT)*mean
  float hs = 0.0f;
  for (int k = 0; k < K_; ++k) hs += hist[k];
  float hd = fmaxf(hs, 1e-8f);
  float unif = 0.0f;
  const float logu = logf(1.0f / (float)K_ + 1e-8f);
  for (int k = 0; k < K_; ++k) {
    float p = hist[k] / hd;
    unif += p * (logf(p + 1e-8f) - logu);
  }
  float num = 0.0f, den = 0.0f;
  for (int r = 0; r < B_ * L_; ++r) {
    float vf = (float)amask[r];
    num += (1.0f - Z[(size_t)r * (K_ + 1) + K_]) * vf;
    den += vf;
  }
  float ratio = num / fmaxf(den, 1.0f);
  float dl = ratio - 0.3f;
  *out_total = rec + cm + 2.0f * unif + 0.1f * (dl * dl);
}

// ------------------------- host orchestration ------------------------
static inline void gemm(hipStream_t st, const float* A, int lda,
                        const float* Bm, int ldb, float* C, int ldc,
                        const float* bias, const float* aux,
                        int M, int N, int Kd, int epi) {
  dim3 g((M + BM_ - 1) / BM_, (N + BN_ - 1) / BN_);
  k_gemm<<<g, 256, 0, st>>>(A, lda, Bm, ldb, C, ldc, bias, aux, M, N, Kd, epi);
}

static void run_layer(hipStream_t st, int layer, float* h, int M, int nSeq,
                      const int* lens, const int* pos, float* XN, float* Q,
                      float* Kb, float* Vb, float* FFb, const float* Wq,
                      const float* Wk, const float* Wv, const float* Wo,
                      const float* Wg, const float* Wu, const float* Wd,
                      const float* g1, const float* g2) {
  const size_t oHH = (size_t)layer * H_ * H_;
  const size_t oHF = (size_t)layer * H_ * FF_;
  const size_t oFH = (size_t)layer * FF_ * H_;
  const size_t oH  = (size_t)layer * H_;
  k_rmsnorm<<<M, 256, 0, st>>>(h, g1 + oH, XN);
  gemm(st, XN, H_, Wq + oHH, H_, Q, H_, nullptr, nullptr, M, H_, H_, EPI_NONE);
  gemm(st, XN, H_, Wk + oHH, H_, Kb, H_, nullptr, nullptr, M, H_, H_, EPI_NONE);
  gemm(st, XN, H_, Wv + oHH, H_, Vb, H_, nullptr, nullptr, M, H_, H_, EPI_NONE);
  k_rope<<<M, 384, 0, st>>>(Q, pos);
  k_rope<<<M, 384, 0, st>>>(Kb, pos);
  k_attn<<<dim3(nSeq, NH_), 96, 0, st>>>(Q, Kb, Vb, XN, lens);  // O -> XN
  gemm(st, XN, H_, Wo + oHH, H_, h, H_, nullptr, h, M, H_, H_, EPI_ADD);
  k_rmsnorm<<<M, 256, 0, st>>>(h, g2 + oH, XN);
  gemm(st, XN, H_, Wg + oHF, FF_, FFb, FF_, nullptr, nullptr, M, FF_, H_, EPI_NONE);
  gemm(st, XN, H_, Wu + oHF, FF_, FFb, FF_, nullptr, FFb, M, FF_, H_, EPI_SILU_MUL);
  gemm(st, FFb, FF_, Wd + oFH, H_, h, H_, nullptr, h, M, H_, FF_, EPI_ADD);
}

extern "C" void kernel_launch(void* const* d_in, const int* in_sizes, int n_in,
                              void* d_out, int out_size, void* d_ws,
                              size_t ws_size, hipStream_t stream) {
  (void)in_sizes; (void)n_in; (void)out_size; (void)ws_size;
  const int*   ids   = (const int*)d_in[0];
  const int*   amask = (const int*)d_in[1];
  const float* gum   = (const float*)d_in[2];
  const float* emb   = (const float*)d_in[3];
  const float* Wq    = (const float*)d_in[4];
  const float* Wk    = (const float*)d_in[5];
  const float* Wv    = (const float*)d_in[6];
  const float* Wo    = (const float*)d_in[7];
  const float* Wg    = (const float*)d_in[8];
  const float* Wu    = (const float*)d_in[9];
  const float* Wd    = (const float*)d_in[10];
  const float* ln1   = (const float*)d_in[11];
  const float* ln2   = (const float*)d_in[12];
  const float* flw   = (const float*)d_in[13];
  const float* lmh   = (const float*)d_in[14];
  const float* chw   = (const float*)d_in[15];
  const float* chb   = (const float*)d_in[16];
  const float* cemb  = (const float*)d_in[17];
  const float* tw1   = (const float*)d_in[18];
  const float* tb1   = (const float*)d_in[19];
  const float* tw2   = (const float*)d_in[20];
  const float* tb2   = (const float*)d_in[21];

  // -------- workspace carve (floats, then ints) --------
  float* W = (float*)d_ws;
  size_t o = 0;
  const size_t RBIG = (size_t)N_ * SMAX_ * H_;     // 18240*768
  float* Hbig = W + o; o += RBIG;
  float* XN   = W + o; o += RBIG;
  float* Qb   = W + o; o += RBIG;
  float* Kbuf = W + o; o += RBIG;
  float* Vbuf = W + o; o += RBIG;
  float* FFb  = W + o; o += (size_t)N_ * SMAX_ * FF_;
  float* hs   = W + o; o += (size_t)B_ * L_ * H_;  // h_shallow (persists)
  float* hm   = W + o; o += (size_t)B_ * L_ * H_;  // c_in -> c_mid
  float* tailh = W + o; o += (size_t)B_ * L_ * 2 * H_;
  float* tailo = W + o; o += (size_t)B_ * L_ * H_;
  float* logC  = W + o; o += (size_t)B_ * L_ * (K_ + 1);
  float* zsoft = W + o; o += (size_t)B_ * L_ * (K_ + 1);
  float* esoft = W + o; o += (size_t)B_ * L_ * H_;
  float* lasth = W + o; o += (size_t)N_ * H_;
  float* hist  = W + o; o += K_;
  float* commit_rows = W + o; o += B_ * L_;
  float* rowloss = W + o; o += N_;
  float* activeF = W + o; o += N_;
  int* I = (int*)(W + o);
  size_t oi = 0;
  int* pos_sh  = I + oi; oi += B_ * L_;
  int* chosen  = I + oi; oi += B_ * L_;
  int* iscomp  = I + oi; oi += B_ * L_;
  int* cpos    = I + oi; oi += B_ * L_;
  int* cum     = I + oi; oi += B_ * L_;
  int* latest  = I + oi; oi += B_ * L_;
  int* counts  = I + oi; oi += B_;
  int* lengths = I + oi; oi += B_;
  int* seqpos  = I + oi; oi += N_ * SMAX_;
  int* seqlen  = I + oi; oi += N_;
  int* lastidx = I + oi; oi += N_;
  int* activeI = I + oi; oi += N_;
  int* targets = I + oi; oi += N_;

  float* logits = (float*)d_out;                  // 190 x 32000
  float* total  = logits + (size_t)N_ * V_;

  const int MS = B_ * L_;                         // 192 shallow/mid rows
  const int MD = N_ * SMAX_;                      // 18240 deep rows

  // -------- shallow stage --------
  k_init<<<1, 256, 0, stream>>>(amask, lengths, pos_sh);
  k_embed<<<MS, 256, 0, stream>>>(ids, emb, hs);
  for (int l = 0; l < 2; ++l)
    run_layer(stream, l, hs, MS, B_, lengths, pos_sh, XN, Qb, Kbuf, Vbuf, FFb,
              Wq, Wk, Wv, Wo, Wg, Wu, Wd, ln1, ln2);

  // -------- concept selection --------
  gemm(stream, hs, H_, chw, K_ + 1, logC, K_ + 1, chb, nullptr,
       MS, K_ + 1, H_, EPI_BIAS);
  k_concept<<<MS, 256, 0, stream>>>(logC, gum, amask, zsoft, chosen, iscomp);
  gemm(stream, zsoft, K_ + 1, cemb, H_, esoft, H_, nullptr, nullptr,
       MS, H_, K_, EPI_NONE);                     // e_soft = w_soft @ concept_emb
  k_commit<<<MS, 256, 0, stream>>>(esoft, chosen, iscomp, cemb, commit_rows);
  k_hist<<<2, 256, 0, stream>>>(zsoft, amask, hist);
  k_scan<<<1, 64, 0, stream>>>(iscomp, counts, cpos, cum, latest);

  // -------- mid stage on compacted concepts --------
  k_cin<<<MS, 256, 0, stream>>>(counts, cpos, chosen, cemb, hm);
  for (int l = 2; l < 4; ++l)
    run_layer(stream, l, hm, MS, B_, counts, cpos, XN, Qb, Kbuf, Vbuf, FFb,
              Wq, Wk, Wv, Wo, Wg, Wu, Wd, ln1, ln2);
  k_maskmid<<<MS, 256, 0, stream>>>(hm, counts);

  // -------- tail MLP on the 192 unique h_shallow rows --------
  gemm(stream, hs, H_, tw1, 2 * H_, tailh, 2 * H_, tb1, nullptr,
       MS, 2 * H_, H_, EPI_GELU_BIAS);
  gemm(stream, tailh, 2 * H_, tw2, H_, tailo, H_, tb2, hs,
       MS, H_, 2 * H_, EPI_BIAS_ADD);

  // -------- assemble deep sequences & run deep stage --------
  k_assemble<<<MD, 256, 0, stream>>>(lengths, cum, latest, cpos, hm, tailo,
                                     ids, Hbig, seqpos, seqlen, lastidx,
                                     activeI, activeF, targets);
  for (int l = 4; l < 6; ++l)
    run_layer(stream, l, Hbig, MD, N_, seqlen, seqpos, XN, Qb, Kbuf, Vbuf, FFb,
              Wq, Wk, Wv, Wo, Wg, Wu, Wd, ln1, ln2);
  k_rmsnorm<<<MD, 256, 0, stream>>>(Hbig, flw, XN);
  k_gather_last<<<N_, 256, 0, stream>>>(XN, lastidx, lasth);

  // -------- logits + losses --------
  gemm(stream, lasth, H_, lmh, V_, logits, V_, nullptr, activeF,
       N_, V_, H_, EPI_ROWSCALE);
  k_rowloss<<<N_, 256, 0, stream>>>(logits, targets, activeI, rowloss);
  k_total<<<1, 32, 0, stream>>>(rowloss, activeI, commit_rows, hist, zsoft,
                                amask, total);
}